// LongTermMemory_8581344657476
// MI455X (gfx1250) — compile-verified
//
#include <hip/hip_runtime.h>

typedef __attribute__((ext_vector_type(2))) float v2f;
typedef __attribute__((ext_vector_type(8))) float v8f;

#define D 256
#define INV_SQRT_D 0.0625f
#define NBLK 256  // blocks per bank for the scoring pass

// ---------- kernel 1: qm = mean(query, axis=0); q = Wq @ qm + bq ----------
__global__ __launch_bounds__(256) void proj_kernel(const float* __restrict__ query,
                                                   const float* __restrict__ Wq,
                                                   const float* __restrict__ bq,
                                                   float* __restrict__ ws) {
  __shared__ float s_qm[D];
  const int t = threadIdx.x;
  float acc = 0.0f;
  for (int r = 0; r < 128; ++r) acc += query[r * D + t];
  const float qm = acc * (1.0f / 128.0f);
  s_qm[t] = qm;
  ws[t] = qm;                 // ws[0..255] = qm
  __syncthreads();
  float a = bq[t];
  #pragma unroll 8
  for (int i = 0; i < D; ++i) a += Wq[t * D + i] * s_qm[i];
  ws[D + t] = a;              // ws[256..511] = q
}

// sorted-desc insert into a 5-element list
__device__ __forceinline__ void topk_insert(float (&ts)[5], int (&ti)[5], float sc, int id) {
  if (sc > ts[4]) {
    ts[4] = sc; ti[4] = id;
    #pragma unroll
    for (int j = 4; j > 0; --j) {
      if (ts[j] > ts[j - 1]) {
        float tf = ts[j]; ts[j] = ts[j - 1]; ts[j - 1] = tf;
        int tn = ti[j]; ti[j] = ti[j - 1]; ti[j - 1] = tn;
      }
    }
  }
}

// ---------- kernel 2: streaming WMMA scores + fused hierarchical top-5 ----------
// Each wave processes PAIRS of 16-key tiles (32 keys) sharing one B-fragment
// stream; lane L owns key row0+L, so all 32 lanes do top-k bookkeeping.
__global__ __launch_bounds__(256) void score_topk_kernel(
    const float* __restrict__ fk, const float* __restrict__ pk, const float* __restrict__ ek,
    const float* __restrict__ ws_q, float* __restrict__ part_s, int* __restrict__ part_i,
    int nf, int np, int ne) {
  __shared__ float s_q[D];
  __shared__ float s_sc[256 * 5];
  __shared__ int   s_ix[256 * 5];

  const int bank = blockIdx.y;
  const float* keys = (bank == 0) ? fk : ((bank == 1) ? pk : ek);
  const int N = (bank == 0) ? nf : ((bank == 1) ? np : ne);

  const int t = threadIdx.x;
  s_q[t] = ws_q[t];
  __syncthreads();

  const int lane = t & 31;
  const int wave = t >> 5;
  const int gwave = blockIdx.x * 8 + wave;
  const int nwaves = gridDim.x * 8;
  const int npairs = N >> 5;                 // all bank sizes divisible by 32
  const int h2 = (lane >> 4) << 1;           // 0 for lanes 0-15, 2 for lanes 16-31
  const int rowl = lane & 15;

  float ts[5]; int ti[5];
  #pragma unroll
  for (int j = 0; j < 5; ++j) { ts[j] = -1e30f; ti[j] = 0; }

  for (int pair = gwave; pair < npairs; pair += nwaves) {
    const int row0 = pair << 5;              // 32 keys: rows row0 .. row0+31
    const float* kp0 = keys + (size_t)(row0 + rowl) * D + h2;
    const float* kp1 = keys + (size_t)(row0 + 16 + rowl) * D + h2;
    const int npair = pair + nwaves;
    if (npair < npairs)
      __builtin_prefetch(keys + (size_t)((npair << 5) + rowl) * D + h2, 0, 1);

    v8f c0 = {0.f, 0.f, 0.f, 0.f, 0.f, 0.f, 0.f, 0.f};
    v8f c1 = {0.f, 0.f, 0.f, 0.f, 0.f, 0.f, 0.f, 0.f};
    #pragma unroll 4
    for (int k = 0; k < D; k += 4) {
      // B (4x16 f32): all 16 columns carry the q chunk; shared by both tiles
      v2f b = *(const v2f*)(s_q + k + h2);
      // A (16x4 f32): lane holds {key[row][k+2h], key[row][k+1+2h]}
      v2f a0 = *(const v2f*)(kp0 + k);
      v2f a1 = *(const v2f*)(kp1 + k);
      c0 = __builtin_amdgcn_wmma_f32_16x16x4_f32(false, a0, false, b, (short)0, c0,
                                                 false, false);
      c1 = __builtin_amdgcn_wmma_f32_16x16x4_f32(false, a1, false, b, (short)0, c1,
                                                 false, false);
    }
    // C layout (each tile): VGPR r lanes0-15 = score(row r), lanes16-31 = score(row r+8).
    // Lane L claims score of key row0+L: tile0 for L<16, tile1 for L>=16.
    const bool hi = lane >= 16;
    const float e0 = hi ? c1[0] : c0[0];
    const float e1 = hi ? c1[1] : c0[1];
    const float e2 = hi ? c1[2] : c0[2];
    const float e3 = hi ? c1[3] : c0[3];
    const float e4 = hi ? c1[4] : c0[4];
    const float e5 = hi ? c1[5] : c0[5];
    const float e6 = hi ? c1[6] : c0[6];
    const float e7 = hi ? c1[7] : c0[7];
    const int cs = lane & 7;
    float v = (cs & 4) ? ((cs & 2) ? ((cs & 1) ? e7 : e6) : ((cs & 1) ? e5 : e4))
                       : ((cs & 2) ? ((cs & 1) ? e3 : e2) : ((cs & 1) ? e1 : e0));
    const int srcLane = (((lane & 8) ? lane + 8 : lane) - (lane & 16));
    float s = __shfl(v, srcLane, 32);
    topk_insert(ts, ti, s * INV_SQRT_D, row0 + lane);
  }

  // block-level tree merge of 256 sorted top-5 lists
  #pragma unroll
  for (int j = 0; j < 5; ++j) { s_sc[t * 5 + j] = ts[j]; s_ix[t * 5 + j] = ti[j]; }
  __syncthreads();
  for (int stride = 128; stride > 0; stride >>= 1) {
    float o[5]; int oi[5];
    if (t < stride) {
      const int pa = t * 5, pb = (t + stride) * 5;
      int ia = 0, ib = 0;
      #pragma unroll
      for (int j = 0; j < 5; ++j) {
        float av = s_sc[pa + ia], bv = s_sc[pb + ib];
        bool ta = av >= bv;
        o[j] = ta ? av : bv;
        oi[j] = ta ? s_ix[pa + ia] : s_ix[pb + ib];
        ia += ta ? 1 : 0; ib += ta ? 0 : 1;
      }
    }
    __syncthreads();
    if (t < stride) {
      #pragma unroll
      for (int j = 0; j < 5; ++j) { s_sc[t * 5 + j] = o[j]; s_ix[t * 5 + j] = oi[j]; }
    }
    __syncthreads();
  }
  if (t == 0) {
    const int base = (bank * gridDim.x + blockIdx.x) * 5;
    #pragma unroll
    for (int j = 0; j < 5; ++j) { part_s[base + j] = s_sc[j]; part_i[base + j] = s_ix[j]; }
  }
}

// ---------- kernel 3: merge per-block partials -> 11 candidates ----------
__global__ __launch_bounds__(256) void merge_kernel(const float* __restrict__ part_s,
                                                    const int* __restrict__ part_i,
                                                    float* __restrict__ cand_s,
                                                    int* __restrict__ cand_i) {
  __shared__ float s_sc[256 * 5];
  __shared__ int   s_ix[256 * 5];
  const int bank = blockIdx.x;
  const int t = threadIdx.x;
  const int base = (bank * NBLK + t) * 5;
  #pragma unroll
  for (int j = 0; j < 5; ++j) { s_sc[t * 5 + j] = part_s[base + j]; s_ix[t * 5 + j] = part_i[base + j]; }
  __syncthreads();
  for (int stride = 128; stride > 0; stride >>= 1) {
    float o[5]; int oi[5];
    if (t < stride) {
      const int pa = t * 5, pb = (t + stride) * 5;
      int ia = 0, ib = 0;
      #pragma unroll
      for (int j = 0; j < 5; ++j) {
        float av = s_sc[pa + ia], bv = s_sc[pb + ib];
        bool ta = av >= bv;
        o[j] = ta ? av : bv;
        oi[j] = ta ? s_ix[pa + ia] : s_ix[pb + ib];
        ia += ta ? 1 : 0; ib += ta ? 0 : 1;
      }
    }
    __syncthreads();
    if (t < stride) {
      #pragma unroll
      for (int j = 0; j < 5; ++j) { s_sc[t * 5 + j] = o[j]; s_ix[t * 5 + j] = oi[j]; }
    }
    __syncthreads();
  }
  if (t == 0) {
    const int off = (bank == 0) ? 0 : ((bank == 1) ? 5 : 8);
    const int kb = (bank == 0) ? 5 : 3;
    for (int j = 0; j < kb; ++j) { cand_s[off + j] = s_sc[j]; cand_i[off + j] = s_ix[j]; }
  }
}

// ---------- kernel 4: rerank 11 candidates, emit top-5 values + scores ----------
__global__ __launch_bounds__(256) void rerank_kernel(
    const float* __restrict__ fv, const float* __restrict__ pv, const float* __restrict__ ev,
    const float* __restrict__ Wr, const float* __restrict__ br_, const float* __restrict__ ws_qm,
    const float* __restrict__ cand_s, const int* __restrict__ cand_i, float* __restrict__ out) {
  __shared__ float red[256];
  __shared__ float s_r[16];
  __shared__ int s_sel[8];
  const int t = threadIdx.x;

  red[t] = ws_qm[t] * Wr[t];
  __syncthreads();
  for (int s2 = 128; s2 > 0; s2 >>= 1) { if (t < s2) red[t] += red[t + s2]; __syncthreads(); }
  const float qdot = red[0];
  __syncthreads();

  for (int c = 0; c < 11; ++c) {
    const float* vb = (c < 5) ? fv : ((c < 8) ? pv : ev);
    const float* vp = vb + (size_t)cand_i[c] * D;
    red[t] = vp[t] * Wr[D + t];
    __syncthreads();
    for (int s2 = 128; s2 > 0; s2 >>= 1) { if (t < s2) red[t] += red[t + s2]; __syncthreads(); }
    if (t == 0) s_r[c] = qdot + red[0] + br_[0];
    __syncthreads();
  }

  if (t == 0) {
    unsigned used = 0;
    for (int j = 0; j < 5; ++j) {
      float best = -1e38f; int bi = 0;
      for (int c = 0; c < 11; ++c)
        if (!((used >> c) & 1u) && s_r[c] > best) { best = s_r[c]; bi = c; }
      used |= (1u << bi);
      s_sel[j] = bi;
    }
  }
  __syncthreads();

  for (int j = 0; j < 5; ++j) {
    const int c = s_sel[j];
    const float* vb = (c < 5) ? fv : ((c < 8) ? pv : ev);
    const float* vp = vb + (size_t)cand_i[c] * D;
    out[j * D + t] = vp[t];
  }
  if (t < 5) out[5 * D + t] = cand_s[s_sel[t]];
}

extern "C" void kernel_launch(void* const* d_in, const int* in_sizes, int n_in,
                              void* d_out, int out_size, void* d_ws, size_t ws_size,
                              hipStream_t stream) {
  const float* query = (const float*)d_in[0];
  const float* fk = (const float*)d_in[1];
  const float* fv = (const float*)d_in[2];
  const float* pk = (const float*)d_in[3];
  const float* pv = (const float*)d_in[4];
  const float* ek = (const float*)d_in[5];
  const float* ev = (const float*)d_in[6];
  const float* Wq = (const float*)d_in[7];
  const float* bq = (const float*)d_in[8];
  const float* Wr = (const float*)d_in[9];
  const float* br = (const float*)d_in[10];
  const int nf = in_sizes[1] / D;
  const int np = in_sizes[3] / D;
  const int ne = in_sizes[5] / D;

  float* ws_f = (float*)d_ws;
  float* ws_qm = ws_f;                         // 256
  float* ws_q = ws_f + D;                      // 256
  float* part_s = ws_f + 2 * D;                // 3*NBLK*5
  int* part_i = (int*)(part_s + 3 * NBLK * 5); // 3*NBLK*5
  float* cand_s = (float*)(part_i + 3 * NBLK * 5); // 11 (+pad)
  int* cand_i = (int*)(cand_s + 16);               // 11

  proj_kernel<<<1, 256, 0, stream>>>(query, Wq, bq, ws_f);
  dim3 grid(NBLK, 3);
  score_topk_kernel<<<grid, 256, 0, stream>>>(fk, pk, ek, ws_q, part_s, part_i, nf, np, ne);
  merge_kernel<<<3, 256, 0, stream>>>(part_s, part_i, cand_s, cand_i);
  rerank_kernel<<<1, 256, 0, stream>>>(fv, pv, ev, Wr, br, ws_qm, cand_s, cand_i,
                                       (float*)d_out);
}